// GraphAttentionLayer_13924283973765
// MI455X (gfx1250) — compile-verified
//
#include <hip/hip_runtime.h>
#include <hip/hip_bf16.h>

// ---------------------------------------------------------------------------
// GAT layer for MI455X (gfx1250, wave32).
//   h = x @ W                      -> fp32 WMMA (v_wmma_f32_16x16x4_f32)
//   scores, edge softmax, scatter  -> bandwidth/atomic bound (HW f32 atomics)
// ---------------------------------------------------------------------------

#define N_NODES 50000
#define N_EDGES 800000
#define D 128
#define NEG_SLOPE 0.2f
#define EPS 1e-8f

typedef __attribute__((ext_vector_type(2))) float v2f;
typedef __attribute__((ext_vector_type(8))) float v8f;

// ---------------------------------------------------------------------------
// Kernel 1: h = x @ W using V_WMMA_F32_16X16X4_F32 (full fp32 precision).
// Block = 256 threads = 8 waves. Block covers 128 rows; each wave owns a
// 16-row strip and sweeps all 8 column tiles, accumulating K=128 in 32 WMMA
// steps. W (64KB) and the 128x128 x-tile (64KB) are staged in LDS (128KB of
// the WGP's 320KB).
// A layout (16x4 f32): lanes 0-15 -> K=0,1 ; lanes 16-31 -> K=2,3 (M = lane&15)
// B layout (4x16 f32): mirrored with N = lane&15
// C/D layout (16x16 f32): VGPR r, lane -> M = r + 8*(lane>>4), N = lane&15
// ---------------------------------------------------------------------------
__global__ __launch_bounds__(256) void gat_gemm_wmma(
    const float* __restrict__ x, const float* __restrict__ W,
    float* __restrict__ h, int n_nodes)
{
    __shared__ __align__(16) float sW[D * D];      // 64 KB
    __shared__ __align__(16) float sA[128 * D];    // 64 KB

    const int tid  = threadIdx.x;
    const int wave = tid >> 5;
    const int lane = tid & 31;
    const int half = lane >> 4;
    const int m    = lane & 15;

    const int rowBase = blockIdx.x * 128;

    // Stage W: 16384 floats with float4 (b128) loads/stores.
    {
        const float4* Wv  = (const float4*)W;
        float4*       sWv = (float4*)sW;
        for (int i = tid; i < (D * D) / 4; i += 256) sWv[i] = Wv[i];
    }
    // Stage x rows [rowBase, rowBase+128), zero-padding past n_nodes.
    {
        const float4* Xv  = (const float4*)x;
        float4*       sAv = (float4*)sA;
        for (int i = tid; i < 128 * (D / 4); i += 256) {
            int r  = i >> 5;        // row within tile
            int c4 = i & 31;        // float4 column
            int gr = rowBase + r;
            float4 v = make_float4(0.f, 0.f, 0.f, 0.f);
            if (gr < n_nodes) v = Xv[(size_t)gr * (D / 4) + c4];
            sAv[i] = v;
        }
    }
    __syncthreads();

    const int row0 = rowBase + wave * 16;
    if (row0 >= n_nodes) return;   // wave-uniform: EXEC stays all-ones for WMMA
                                   // (N_NODES % 16 == 0, so active strips are full)

    const float* Arow = &sA[(size_t)(wave * 16 + m) * D];

    #pragma unroll
    for (int ct = 0; ct < 8; ++ct) {
        const int col0 = ct * 16;
        v8f c = {};
        #pragma unroll
        for (int kk = 0; kk < 32; ++kk) {
            const int k0 = kk * 4 + half * 2;
            v2f a, b;
            a[0] = Arow[k0];
            a[1] = Arow[k0 + 1];
            b[0] = sW[(size_t)k0 * D + col0 + m];
            b[1] = sW[(size_t)(k0 + 1) * D + col0 + m];
            c = __builtin_amdgcn_wmma_f32_16x16x4_f32(
                    /*neg_a=*/false, a, /*neg_b=*/false, b,
                    /*c_mod=*/(short)0, c, /*reuse_a=*/false, /*reuse_b=*/false);
        }
        float* hrow = &h[(size_t)(row0 + half * 8) * D + col0 + m];
        #pragma unroll
        for (int r = 0; r < 8; ++r) hrow[(size_t)r * D] = c[r];
    }
}

// ---------------------------------------------------------------------------
// Kernel 2: s_src[i] = h[i,:]·a[:128], s_tgt[i] = h[i,:]·a[128:].
// One wave per node; float4 loads; __shfl_xor tree reduction (wave32).
// ---------------------------------------------------------------------------
__global__ __launch_bounds__(256) void gat_scores(
    const float* __restrict__ h, const float* __restrict__ a,
    float* __restrict__ s_src, float* __restrict__ s_tgt, int n_nodes)
{
    const int wave = threadIdx.x >> 5;
    const int lane = threadIdx.x & 31;
    const int node = blockIdx.x * 8 + wave;
    if (node >= n_nodes) return;

    const float4 hv = ((const float4*)(h + (size_t)node * D))[lane];
    const float4 as = ((const float4*)a)[lane];
    const float4 at = ((const float4*)(a + D))[lane];

    float ps = hv.x * as.x + hv.y * as.y + hv.z * as.z + hv.w * as.w;
    float pt = hv.x * at.x + hv.y * at.y + hv.z * at.z + hv.w * at.w;
    #pragma unroll
    for (int off = 16; off > 0; off >>= 1) {
        ps += __shfl_xor(ps, off, 32);
        pt += __shfl_xor(pt, off, 32);
    }
    if (lane == 0) { s_src[node] = ps; s_tgt[node] = pt; }
}

// ---------------------------------------------------------------------------
// Kernel 3: per-edge softmax numerator + denominator scatter.
// exp_e[e] = exp(leaky_relu(s_src[src]+s_tgt[tgt]) * w[e]);
// denom[tgt] += exp_e[e]  (HW global_atomic_add_f32, L2-resident target).
// ---------------------------------------------------------------------------
__global__ __launch_bounds__(256) void gat_edge_num(
    const int* __restrict__ ei, const float* __restrict__ ew,
    const float* __restrict__ s_src, const float* __restrict__ s_tgt,
    float* __restrict__ exp_e, float* __restrict__ denom, int n_edges)
{
    const int e = blockIdx.x * blockDim.x + threadIdx.x;
    if (e >= n_edges) return;
    const int src = ei[e];
    const int tgt = ei[n_edges + e];
    float s = s_src[src] + s_tgt[tgt];
    s = (s > 0.f) ? s : s * NEG_SLOPE;
    const float v = expf(s * ew[e]);
    exp_e[e] = v;
    unsafeAtomicAdd(&denom[tgt], v);
}

// ---------------------------------------------------------------------------
// Kernel 4: h_prime[tgt,:] += alpha_e * h[src,:].  One wave per edge: each
// lane gathers a float4 of h[src] and issues 4 f32 atomics into out (25.6 MB
// accumulator lives in the 192 MB L2). This stage dominates runtime.
// ---------------------------------------------------------------------------
__global__ __launch_bounds__(256) void gat_aggregate(
    const int* __restrict__ ei, const float* __restrict__ exp_e,
    const float* __restrict__ denom, const float* __restrict__ h,
    float* __restrict__ out, int n_edges)
{
    const int wave = threadIdx.x >> 5;
    const int lane = threadIdx.x & 31;
    const int e = blockIdx.x * 8 + wave;
    if (e >= n_edges) return;
    const int src = ei[e];
    const int tgt = ei[n_edges + e];
    const float alpha = exp_e[e] / (denom[tgt] + EPS);
    const float4 hv = ((const float4*)(h + (size_t)src * D))[lane];
    float* o = out + (size_t)tgt * D + lane * 4;
    unsafeAtomicAdd(o + 0, hv.x * alpha);
    unsafeAtomicAdd(o + 1, hv.y * alpha);
    unsafeAtomicAdd(o + 2, hv.z * alpha);
    unsafeAtomicAdd(o + 3, hv.w * alpha);
}

// ---------------------------------------------------------------------------
// Kernel 5: in-place ELU on the accumulated output.
// ---------------------------------------------------------------------------
__global__ __launch_bounds__(256) void gat_elu(float* __restrict__ out, int n)
{
    const int i = blockIdx.x * blockDim.x + threadIdx.x;
    if (i < n) {
        const float v = out[i];
        out[i] = (v > 0.f) ? v : expm1f(v);
    }
}

// ---------------------------------------------------------------------------
// Launch. Inputs (setup_inputs order): x, edge_index, edge_weights, W, a.
// d_out is used directly as the h' accumulator (zeroed each call), saving
// 25.6 MB of workspace. Workspace layout (floats):
//   h[50000*128] | s_src[50000] | s_tgt[50000] | denom[50000] | exp_e[800000]
// ---------------------------------------------------------------------------
extern "C" void kernel_launch(void* const* d_in, const int* in_sizes, int n_in,
                              void* d_out, int out_size, void* d_ws, size_t ws_size,
                              hipStream_t stream)
{
    const float* x  = (const float*)d_in[0];
    const int*   ei = (const int*)  d_in[1];
    const float* ew = (const float*)d_in[2];
    const float* W  = (const float*)d_in[3];
    const float* a  = (const float*)d_in[4];
    float* out = (float*)d_out;

    float* ws     = (float*)d_ws;
    float* h      = ws;                       // 6,400,000 floats
    float* s_src  = h + (size_t)N_NODES * D;  // 50,000
    float* s_tgt  = s_src + N_NODES;          // 50,000
    float* denom  = s_tgt + N_NODES;          // 50,000
    float* exp_e  = denom + N_NODES;          // 800,000

    hipMemsetAsync(out,   0, sizeof(float) * (size_t)N_NODES * D, stream);
    hipMemsetAsync(denom, 0, sizeof(float) * N_NODES, stream);

    gat_gemm_wmma<<<(N_NODES + 127) / 128, 256, 0, stream>>>(x, W, h, N_NODES);
    gat_scores  <<<(N_NODES + 7) / 8,      256, 0, stream>>>(h, a, s_src, s_tgt, N_NODES);
    gat_edge_num<<<(N_EDGES + 255) / 256,  256, 0, stream>>>(ei, ew, s_src, s_tgt,
                                                             exp_e, denom, N_EDGES);
    gat_aggregate<<<(N_EDGES + 7) / 8,     256, 0, stream>>>(ei, exp_e, denom, h,
                                                             out, N_EDGES);
    gat_elu     <<<((N_NODES * D) + 255) / 256, 256, 0, stream>>>(out, N_NODES * D);
}